// RelGraphConvLayer_1675037246051
// MI455X (gfx1250) — compile-verified
//
#include <hip/hip_runtime.h>

// ---------------- problem constants (match reference) ----------------
#define N_NODES 30000
#define N_REL   8
#define N_EDGES 131072
#define FEAT    512                  // IN_FEAT == OUT_FEAT == 512
#define MB      64                   // M rows per block (4 m-tiles of 16)
#define M_BLKS  ((N_NODES + MB - 1) / MB)   // 469 (last block partial)

typedef __attribute__((ext_vector_type(2))) float v2f;
typedef __attribute__((ext_vector_type(8))) float v8f;

__device__ __forceinline__ void atomAdd(float* p, float v) {
    // relaxed, agent scope, result unused -> native global_atomic_add_f32
    __hip_atomic_fetch_add(p, v, __ATOMIC_RELAXED, __HIP_MEMORY_SCOPE_AGENT);
}

// ---------------------------------------------------------------------
// zero a float4 region (grid-stride)
__global__ void zero_f4(float4* __restrict__ p, int n4) {
    int i = blockIdx.x * blockDim.x + threadIdx.x;
    int stride = gridDim.x * blockDim.x;
    float4 z; z.x = 0.f; z.y = 0.f; z.z = 0.f; z.w = 0.f;
    for (; i < n4; i += stride) p[i] = z;
}

// in-degree count per destination node (float so GEMM can load directly)
__global__ void count_deg(const int* __restrict__ dst, float* __restrict__ deg,
                          int nEdges) {
    int e = blockIdx.x * blockDim.x + threadIdx.x;
    if (e < nEdges) atomAdd(&deg[dst[e]], 1.0f);
}

// scatter-add x[src[e]] row into agg[dst[e]] row.
// 128 threads per edge, one float4 chunk each. x (61MB) and agg (61MB) both
// fit in the 192MB L2 -> gathers and atomics are L2 traffic, not HBM.
__global__ void scatter_edges(const float* __restrict__ x,
                              const int* __restrict__ src,
                              const int* __restrict__ dst,
                              float* __restrict__ agg, int nEdges) {
    int gid   = blockIdx.x * blockDim.x + threadIdx.x;
    int chunk = gid & 127;   // FEAT/4 = 128 chunks
    int e     = gid >> 7;
    if (e >= nEdges) return;
    int s = src[e];
    int d = dst[e];
    float4 v = ((const float4*)(x + (size_t)s * FEAT))[chunk];
    float* o = agg + (size_t)d * FEAT + chunk * 4;
    atomAdd(o + 0, v.x);
    atomAdd(o + 1, v.y);
    atomAdd(o + 2, v.z);
    atomAdd(o + 3, v.w);
}

// ---------------------------------------------------------------------
// out[M,N] (+)= (A / max(deg,1)) @ W   with M=30000, K=N=512, f32 WMMA.
// One wave -> 64(M) x 64(N) output tile: 4 m-tiles x 4 n-tiles, so each A
// fragment feeds 4 WMMAs and each B fragment feeds 4 WMMAs
// (12 load instrs : 16 wmma per K-step). 8 waves/block cover N=512.
// V_WMMA_F32_16X16X4_F32 lane layout (ISA 7.12.2):
//   A (16x4):  lane L holds A[m = L&15][k0 + 2*(L>>4) + {0,1}]       (v2f)
//   B (4x16):  lane L holds B[k0 + 2*(L>>4) + {0,1}][n0 + (L&15)]    (v2f)
//   C/D:       vgpr i, lane L -> row (i + 8*(L>>4)), col n0 + (L&15) (v8f)
__global__ void __launch_bounds__(256)
rgcn_gemm_f32(const float* __restrict__ A, const float* __restrict__ W,
              const float* __restrict__ deg, float* __restrict__ out,
              int accumulate) {
    const int lane  = threadIdx.x & 31;
    const int wave  = threadIdx.x >> 5;
    const int lo    = lane & 15;
    const int hi    = lane >> 4;              // 0 or 1
    const int mBase = blockIdx.x * MB;
    const int nBase = wave * 64;

    // ---- per-lane A rows (4 m-tiles) + degree normalization ----
    const float* aPtr[4];
    float rdeg[4];
    #pragma unroll
    for (int mt = 0; mt < 4; ++mt) {
        int aRow = mBase + mt * 16 + lo;
        if (aRow >= N_NODES) aRow = N_NODES - 1;       // clamp (tail block)
        aPtr[mt] = A + (size_t)aRow * FEAT + hi * 2;   // index by +k0
        float r = 1.0f;
        if (deg != nullptr) r = 1.0f / fmaxf(deg[aRow], 1.0f);
        rdeg[mt] = r;
    }

    // ---- B fragment base pointers (4 n-tiles) ----
    const float* bPtr[4];
    #pragma unroll
    for (int nt = 0; nt < 4; ++nt)
        bPtr[nt] = W + (size_t)(hi * 2) * FEAT + (nBase + nt * 16 + lo);

    // ---- accumulators ----
    v8f c[4][4];
    #pragma unroll
    for (int mt = 0; mt < 4; ++mt)
        #pragma unroll
        for (int nt = 0; nt < 4; ++nt)
            #pragma unroll
            for (int i = 0; i < 8; ++i) c[mt][nt][i] = 0.f;

    if (accumulate) {
        #pragma unroll
        for (int mt = 0; mt < 4; ++mt)
            #pragma unroll
            for (int i = 0; i < 8; ++i) {
                const int row = mBase + mt * 16 + i + hi * 8;
                if (row < N_NODES) {
                    const size_t rb = (size_t)row * FEAT;
                    #pragma unroll
                    for (int nt = 0; nt < 4; ++nt)
                        c[mt][nt][i] = out[rb + nBase + nt * 16 + lo];
                }
            }
    }

    // ---- K loop: 128 steps of 4, 16 WMMA each ----
    #pragma unroll 2
    for (int k0 = 0; k0 < FEAT; k0 += 4) {
        v2f a[4];
        #pragma unroll
        for (int mt = 0; mt < 4; ++mt) {
            a[mt] = *(const v2f*)(aPtr[mt] + k0);
            a[mt].x *= rdeg[mt];
            a[mt].y *= rdeg[mt];
        }
        const size_t kb = (size_t)k0 * FEAT;
        v2f b[4];
        #pragma unroll
        for (int nt = 0; nt < 4; ++nt) {
            b[nt].x = bPtr[nt][kb];
            b[nt].y = bPtr[nt][kb + FEAT];
        }
        #pragma unroll
        for (int mt = 0; mt < 4; ++mt)
            #pragma unroll
            for (int nt = 0; nt < 4; ++nt)
                c[mt][nt] = __builtin_amdgcn_wmma_f32_16x16x4_f32(
                    false, a[mt], false, b[nt], (short)0, c[mt][nt], false, false);
    }

    // ---- store (predicated only for the tail block; WMMA loop already done,
    //       so EXEC divergence here is legal) ----
    #pragma unroll
    for (int mt = 0; mt < 4; ++mt)
        #pragma unroll
        for (int i = 0; i < 8; ++i) {
            const int row = mBase + mt * 16 + i + hi * 8;
            if (row < N_NODES) {
                const size_t rb = (size_t)row * FEAT;
                #pragma unroll
                for (int nt = 0; nt < 4; ++nt)
                    out[rb + nBase + nt * 16 + lo] = c[mt][nt][i];
            }
        }
}

// h = relu(h + bias), float4-vectorized
__global__ void bias_relu(float* __restrict__ h, const float* __restrict__ bias) {
    int i = blockIdx.x * blockDim.x + threadIdx.x;   // 0 .. N_NODES*128-1
    float4 v = ((float4*)h)[i];
    float4 b = ((const float4*)bias)[i & 127];       // col chunk = i % (FEAT/4)
    v.x = fmaxf(v.x + b.x, 0.f);
    v.y = fmaxf(v.y + b.y, 0.f);
    v.z = fmaxf(v.z + b.z, 0.f);
    v.w = fmaxf(v.w + b.w, 0.f);
    ((float4*)h)[i] = v;
}

// ---------------------------------------------------------------------
extern "C" void kernel_launch(void* const* d_in, const int* in_sizes, int n_in,
                              void* d_out, int out_size, void* d_ws, size_t ws_size,
                              hipStream_t stream) {
    const float* x      = (const float*)d_in[0];   // [30000,512]
    const float* weight = (const float*)d_in[1];   // [8,512,512]
    const float* loop_w = (const float*)d_in[2];   // [512,512]
    const float* bias   = (const float*)d_in[3];   // [512]
    const int*   src    = (const int*)d_in[4];     // [8,131072]
    const int*   dst    = (const int*)d_in[5];     // [8,131072]
    float*       out    = (float*)d_out;           // [30000,512]

    // workspace: agg [30000*512] f32 followed by deg [30000] f32  (~61.6 MB)
    float* agg = (float*)d_ws;
    float* deg = agg + (size_t)N_NODES * FEAT;
    const int zeroN4 = (N_NODES * FEAT + N_NODES) / 4;   // agg+deg together

    for (int r = 0; r < N_REL; ++r) {
        const int* src_r = src + (size_t)r * N_EDGES;
        const int* dst_r = dst + (size_t)r * N_EDGES;

        zero_f4<<<2048, 256, 0, stream>>>((float4*)agg, zeroN4);
        count_deg<<<N_EDGES / 256, 256, 0, stream>>>(dst_r, deg, N_EDGES);
        scatter_edges<<<(N_EDGES * (FEAT / 4)) / 256, 256, 0, stream>>>(
            x, src_r, dst_r, agg, N_EDGES);
        rgcn_gemm_f32<<<M_BLKS, 256, 0, stream>>>(
            agg, weight + (size_t)r * FEAT * FEAT, deg, out, r > 0 ? 1 : 0);
    }

    // self-loop GEMM: out += x @ loop_weight (no degree normalization)
    rgcn_gemm_f32<<<M_BLKS, 256, 0, stream>>>(x, loop_w, nullptr, out, 1);

    // bias + relu
    bias_relu<<<(N_NODES * (FEAT / 4)) / 256, 256, 0, stream>>>(out, bias);
}